// PatchNCELoss_42494406427448
// MI455X (gfx1250) — compile-verified
//
#include <hip/hip_runtime.h>
#include <hip/hip_bf16.h>
#include <stdint.h>
#include <math.h>

typedef __attribute__((ext_vector_type(2))) float v2f;
typedef __attribute__((ext_vector_type(4))) float v4f;
typedef __attribute__((ext_vector_type(8))) float v8f;

#define BATCH     4
#define NROWS     16384
#define DDIM      256
#define PPATCH    4096
#define NTILES    (PPATCH / 16)
#define INV_T     14.285714285714286f
#define NEG_MASKV -10.0f
#define KSTRIDE   260   // padded floats per LDS tile row: 4c+4t bank groups, 2-per-bank b128 reads

__device__ __forceinline__ void lse_update(float& m, float& s, float v) {
    float nm = fmaxf(m, v);
    s = s * __expf(m - nm) + __expf(v - nm);
    m = nm;
}

__global__ __launch_bounds__(128)
void patch_nce_fused_kernel(const float* __restrict__ q,
                            const float* __restrict__ k,
                            float* __restrict__ out) {
    __shared__ float ktile[2][16 * KSTRIDE];   // double-buffered K tile

    const int tid    = threadIdx.x;
    const int lane   = tid & 31;
    const int wave   = tid >> 5;
    const int lane15 = lane & 15;
    const int hi     = lane >> 4;

    const int b      = blockIdx.x >> 6;                 // 64 blocks per batch element
    const int rowblk = blockIdx.x & 63;
    const int r0     = (rowblk << 6) + (wave << 4);     // within-batch row base of this wave

    const float* qb = q + (size_t)b * PPATCH * DDIM;
    const float* kb = k + (size_t)b * PPATCH * DDIM;

    // ---- Preload A panel (16 rows x 256), permuted-K WMMA layout ----
    // step kk consumes d = {2kk, 2kk+1} (lanes 0-15, K=0,1) and {128+2kk, 128+2kk+1} (lanes 16-31, K=2,3)
    // => per-lane A data is one contiguous 512B half-row: compiler emits b128 global loads.
    v2f a[64];
    {
        const float* qrow = qb + (size_t)(r0 + lane15) * DDIM + hi * 128;
        #pragma unroll
        for (int kk = 0; kk < 64; ++kk)
            a[kk] = *(const v2f*)(qrow + 2 * kk);
    }

    float m[8], s[8], lp[8];
    #pragma unroll
    for (int i = 0; i < 8; ++i) { m[i] = -INFINITY; s[i] = 0.f; lp[i] = 0.f; }

    const unsigned ldsb0 = (unsigned)(unsigned long long)(uintptr_t)&ktile[0][0];
    const unsigned ldsb1 = (unsigned)(unsigned long long)(uintptr_t)&ktile[1][0];
    const int ct_diag = r0 >> 4;

    // ---- async stage of one 16KB K tile (contiguous slab) into LDS buffer ----
    auto stage = [&](int ct, unsigned ldsbase) {
        const float* kt = kb + (size_t)ct * 16 * DDIM;
        #pragma unroll
        for (int i = 0; i < 8; ++i) {
            int chunk = tid + i * 128;          // 1024 x 16B chunks
            int krow  = chunk >> 6;             // 64 chunks per tile row
            int cb    = (chunk & 63) << 4;      // byte offset within row
            unsigned lds_addr = ldsbase + (unsigned)(krow * (KSTRIDE * 4) + cb);
            unsigned long long ga =
                (unsigned long long)(uintptr_t)(kt + (size_t)krow * DDIM) + (unsigned)cb;
            asm volatile("global_load_async_to_lds_b128 %0, %1, off"
                         :: "v"(lds_addr), "v"(ga) : "memory");
        }
    };

    // Prologue: fill buffer 0 with tile 0.
    stage(0, ldsb0);
#if __has_builtin(__builtin_amdgcn_s_wait_asynccnt)
    __builtin_amdgcn_s_wait_asynccnt(0);
#else
    asm volatile("s_wait_asynccnt 0" ::: "memory");
#endif
    __syncthreads();

    for (int ct = 0; ct < NTILES; ++ct) {
        // Prefetch next tile into the other buffer; overlaps with this tile's compute.
        // (Other buffer's last readers passed the barrier at the end of iteration ct-1.)
        if (ct + 1 < NTILES)
            stage(ct + 1, ((ct + 1) & 1) ? ldsb1 : ldsb0);

        // ---- 16x16 S tile: 64 chained fp32 WMMAs; one ds_load_b128 feeds two WMMAs ----
        v8f c = {0.f, 0.f, 0.f, 0.f, 0.f, 0.f, 0.f, 0.f};
        const float* lrow = &ktile[ct & 1][lane15 * KSTRIDE + hi * 128];
        #pragma unroll
        for (int t = 0; t < 32; ++t) {
            v4f bq = *(const v4f*)(lrow + 4 * t);
            v2f b0 = __builtin_shufflevector(bq, bq, 0, 1);
            v2f b1 = __builtin_shufflevector(bq, bq, 2, 3);
            c = __builtin_amdgcn_wmma_f32_16x16x4_f32(false, a[2 * t],     false, b0,
                                                      (short)0, c, false, false);
            c = __builtin_amdgcn_wmma_f32_16x16x4_f32(false, a[2 * t + 1], false, b1,
                                                      (short)0, c, false, false);
        }

        // ---- Fused online logsumexp over this tile's 16 columns ----
        // C layout: VGPR i -> row i (lanes 0-15, N=lane) / row 8+i (lanes 16-31, N=lane-16)
        const bool dt = (ct == ct_diag);
        #pragma unroll
        for (int i = 0; i < 8; ++i) {
            float v = c[i] * INV_T;
            if (dt && (lane == i || lane == i + 24)) {
                lp[i] = c[i];                                // positive logit (diag)
                lse_update(m[i], s[i], NEG_MASKV * INV_T);   // masked diagonal term
            }
            lse_update(m[i], s[i], v);                       // normal column (or l_pos col)
        }

        // Next tile's copy must be complete and visible to all waves before it is read,
        // and this barrier also protects the buffer we just read from being re-staged early.
#if __has_builtin(__builtin_amdgcn_s_wait_asynccnt)
        __builtin_amdgcn_s_wait_asynccnt(0);
#else
        asm volatile("s_wait_asynccnt 0" ::: "memory");
#endif
        __syncthreads();
    }

    // ---- Merge 16 per-lane partial logsumexps within each half-wave ----
    #pragma unroll
    for (int mask = 1; mask <= 8; mask <<= 1) {
        #pragma unroll
        for (int i = 0; i < 8; ++i) {
            float mo = __shfl_xor(m[i], mask, 32);
            float so = __shfl_xor(s[i], mask, 32);
            float nm = fmaxf(m[i], mo);
            s[i] = s[i] * __expf(m[i] - nm) + so * __expf(mo - nm);
            m[i] = nm;
        }
    }

    // loss = logsumexp - l_pos/T ; rows 0-7 by lanes 0-7, rows 8-15 by lanes 24-31
    float* ob = out + (size_t)b * PPATCH + r0;
    #pragma unroll
    for (int i = 0; i < 8; ++i) {
        if (lane == i)      ob[i]     = m[i] + __logf(s[i]) - lp[i] * INV_T;
        if (lane == i + 24) ob[8 + i] = m[i] + __logf(s[i]) - lp[i] * INV_T;
    }
}

extern "C" void kernel_launch(void* const* d_in, const int* in_sizes, int n_in,
                              void* d_out, int out_size, void* d_ws, size_t ws_size,
                              hipStream_t stream) {
    (void)in_sizes; (void)n_in; (void)d_ws; (void)ws_size; (void)out_size;
    const float* q = (const float*)d_in[0];
    const float* k = (const float*)d_in[1];
    float* out = (float*)d_out;
    dim3 grid(NROWS / 64);   // 4 batches x 64 row-blocks of 64 rows
    dim3 block(128);         // 4 wave32s, 16 rows each
    patch_nce_fused_kernel<<<grid, block, 0, stream>>>(q, k, out);
}